// Model_14946486190730
// MI455X (gfx1250) — compile-verified
//
#include <hip/hip_runtime.h>
#include <math.h>

typedef __attribute__((ext_vector_type(2))) float v2f;
typedef __attribute__((ext_vector_type(4))) float v4f;
typedef __attribute__((ext_vector_type(8))) float v8f;

#define DIM 128
#define ROWS_PER_BLOCK 128
#define LDS_STRIDE 132  // 128 + 4 pad floats: breaks the 512B row stride so the
                        // 16-row fragment reads don't all land on one LDS bank

// out[r,c] (+)= act( A[r,:] @ W[:,c] + bias[c] )
// A:[N,128] row-major, W:[128,128] row-major. One block = 256 thr = 8 wave32.
// Wave w owns columns [16w,16w+16); block processes 128 rows (8 row tiles).
// B fragments for the whole K=128 loop live in registers (32 x v2f = 64 VGPR).
__global__ __launch_bounds__(256)
void sage_gemm128_wmma(const float* __restrict__ A,
                       const float* __restrict__ W,
                       const float* __restrict__ bias,
                       float* __restrict__ out,
                       int N, int do_relu, int do_accum)
{
    __shared__ float lds[ROWS_PER_BLOCK * LDS_STRIDE];
    const int tid  = threadIdx.x;
    const int lane = tid & 31;
    const int wave = tid >> 5;
    const int row0 = blockIdx.x * ROWS_PER_BLOCK;

    // ---- stage A tile (128x128 f32) into LDS, coalesced 16B loads ----
    for (int i = tid; i < ROWS_PER_BLOCK * (DIM / 4); i += 256) {
        int r  = i / (DIM / 4);
        int c4 = i % (DIM / 4);
        int gr = row0 + r;
        if (gr >= N) gr = N - 1;               // clamp: safe garbage, stores are guarded
        v4f v = *(const v4f*)(A + (size_t)gr * DIM + (size_t)c4 * 4);
        float* p = &lds[r * LDS_STRIDE + c4 * 4];
        p[0] = v.x; p[1] = v.y; p[2] = v.z; p[3] = v.w;
    }
    __syncthreads();

    // ---- load this wave's B fragments for all 32 K-steps ----
    // B 4x16 layout: v0 = {K0 | K2}, v1 = {K1 | K3}; lanes 0-15 / 16-31 halves.
    const int col  = wave * 16 + (lane & 15);
    const int ksel = (lane >> 4) * 2;          // 0 for lanes 0-15, 2 for 16-31
    v2f bfrag[32];
    #pragma unroll
    for (int ks = 0; ks < 32; ++ks) {
        int k = ks * 4 + ksel;
        bfrag[ks].x = W[(size_t)k * DIM + col];
        bfrag[ks].y = W[(size_t)(k + 1) * DIM + col];
    }

    const float bv = bias ? bias[col] : 0.0f;
    const int half = (lane >> 4) * 8;          // C/D: lanes 16-31 hold rows g+8

    for (int rt = 0; rt < 8; ++rt) {
        const int rl = rt * 16 + (lane & 15);  // A frag row (both lane halves: M = lane%16)

        // ---- init accumulator: bias, or existing out values for accumulation ----
        v8f c;
        #pragma unroll
        for (int g = 0; g < 8; ++g) {
            if (do_accum) {
                int gr = row0 + rt * 16 + g + half;
                c[g] = (gr < N) ? out[(size_t)gr * DIM + col] : 0.0f;
            } else {
                c[g] = bv;
            }
        }

        // ---- K loop: 32 x v_wmma_f32_16x16x4_f32 ----
        #pragma unroll
        for (int ks = 0; ks < 32; ++ks) {
            // A 16x4 layout: v0 = {K0 | K2}, v1 = {K1 | K3} -> per-lane v2f from LDS
            v2f a = *(const v2f*)&lds[rl * LDS_STRIDE + ks * 4 + ksel];
            c = __builtin_amdgcn_wmma_f32_16x16x4_f32(
                    /*neg_a=*/false, a, /*neg_b=*/false, bfrag[ks],
                    /*c_mod=*/(short)0, c, /*reuse_a=*/false, /*reuse_b=*/false);
        }

        // ---- epilogue: optional relu, guarded store ----
        #pragma unroll
        for (int g = 0; g < 8; ++g) {
            int gr = row0 + rt * 16 + g + half;
            float v = c[g];
            if (do_relu) v = fmaxf(v, 0.0f);
            if (gr < N) out[(size_t)gr * DIM + col] = v;
        }
    }
}

// h_neigh = 0 (uint 0 == float +0.0, the correct identity for max of relu outputs,
// and the reference's value for isolated nodes)
__global__ void sage_fill_zero(unsigned int* __restrict__ p, size_t n)
{
    size_t i = (size_t)blockIdx.x * blockDim.x + threadIdx.x;
    if (i < n) p[i] = 0u;
}

// segment_max via u32 atomicMax (valid: all m >= 0 after relu, IEEE order == uint order).
// One wave per edge; lane handles 4 consecutive floats (float4, fully coalesced).
__global__ __launch_bounds__(256)
void sage_scatter_max(const float* __restrict__ m,
                      const int* __restrict__ src,
                      const int* __restrict__ dst,
                      unsigned int* __restrict__ hn,
                      int E)
{
    int gw   = (int)(((size_t)blockIdx.x * blockDim.x + threadIdx.x) >> 5);
    int lane = threadIdx.x & 31;
    if (gw >= E) return;
    int s = src[gw];
    int d = dst[gw];
    v4f v = *(const v4f*)(m + (size_t)s * DIM + (size_t)lane * 4);
    unsigned int* p = hn + (size_t)d * DIM + (size_t)lane * 4;
    atomicMax(p + 0, __float_as_uint(v.x));
    atomicMax(p + 1, __float_as_uint(v.y));
    atomicMax(p + 2, __float_as_uint(v.z));
    atomicMax(p + 3, __float_as_uint(v.w));
}

// score[e,:] = h[src[e],:] * h[dst[e],:]   (327.7 MB coalesced write; h is L2-hot)
__global__ __launch_bounds__(256)
void sage_edge_score(const float* __restrict__ h,
                     const int* __restrict__ src,
                     const int* __restrict__ dst,
                     float* __restrict__ out,
                     int E)
{
    int gw   = (int)(((size_t)blockIdx.x * blockDim.x + threadIdx.x) >> 5);
    int lane = threadIdx.x & 31;
    if (gw >= E) return;
    int s = src[gw];
    int d = dst[gw];
    v4f a = *(const v4f*)(h + (size_t)s * DIM + (size_t)lane * 4);
    v4f b = *(const v4f*)(h + (size_t)d * DIM + (size_t)lane * 4);
    v4f r; r.x = a.x * b.x; r.y = a.y * b.y; r.z = a.z * b.z; r.w = a.w * b.w;
    *(v4f*)(out + (size_t)gw * DIM + (size_t)lane * 4) = r;
}

extern "C" void kernel_launch(void* const* d_in, const int* in_sizes, int n_in,
                              void* d_out, int out_size, void* d_ws, size_t ws_size,
                              hipStream_t stream)
{
    const float* x       = (const float*)d_in[0];
    const int*   src     = (const int*)  d_in[1];
    const int*   dst     = (const int*)  d_in[2];
    const float* W_pool  = (const float*)d_in[3];
    const float* b_pool  = (const float*)d_in[4];
    const float* W_self  = (const float*)d_in[5];
    const float* W_neigh = (const float*)d_in[6];
    const float* bias    = (const float*)d_in[7];
    float*       score   = (float*)d_out;

    const int N = in_sizes[0] / DIM;
    const int E = in_sizes[1];

    // workspace layout: m | h_neigh | h  (each N*128 f32 = 20.5 MB, L2-resident)
    float* m  = (float*)d_ws;
    float* hn = m  + (size_t)N * DIM;
    float* h  = hn + (size_t)N * DIM;

    const int gblocks = (N + ROWS_PER_BLOCK - 1) / ROWS_PER_BLOCK;
    const int eblocks = (E + 7) / 8;                 // 8 wave32 (= 8 edges) per block
    const size_t nz   = (size_t)N * DIM;

    // m = relu(x @ W_pool + b_pool)
    sage_gemm128_wmma<<<gblocks, 256, 0, stream>>>(x, W_pool, b_pool, m, N, 1, 0);
    // h_neigh = 0; then scatter-max m[src] onto dst
    sage_fill_zero<<<(int)((nz + 255) / 256), 256, 0, stream>>>((unsigned int*)hn, nz);
    sage_scatter_max<<<eblocks, 256, 0, stream>>>(m, src, dst, (unsigned int*)hn, E);
    // h = x @ W_self + bias;  h += h_neigh @ W_neigh
    sage_gemm128_wmma<<<gblocks, 256, 0, stream>>>(x,  W_self,  bias,    h, N, 0, 0);
    sage_gemm128_wmma<<<gblocks, 256, 0, stream>>>(hn, W_neigh, nullptr, h, N, 0, 1);
    // score = h[src] * h[dst]
    sage_edge_score<<<eblocks, 256, 0, stream>>>(h, src, dst, score, E);
}